// FeaturePropagation_47545287967130
// MI455X (gfx1250) — compile-verified
//
#include <hip/hip_runtime.h>
#include <hip/hip_bf16.h>

typedef float v2f __attribute__((ext_vector_type(2)));
typedef float v8f __attribute__((ext_vector_type(8)));
typedef unsigned int v4u __attribute__((ext_vector_type(4)));
typedef int v4i __attribute__((ext_vector_type(4)));
typedef int v8i __attribute__((ext_vector_type(8)));

#define B_     4
#define N1_    16384
#define N2_    8192
#define C_IN   256
#define C_SKIP 64
#define C0_    320
#define C1_    256
#define KT2    2048   // KNN key tile (double-buffered: 2*3*2048*4 = 48KB LDS)

// Detect 6-arg tensor_load_to_lds toolchain (therock headers) vs 5-arg ROCm 7.2.
#if defined(__has_include)
#if __has_include(<hip/amd_detail/amd_gfx1250_TDM.h>)
#define TDM_6ARG 1
#endif
#endif

// ---------------------------------------------------------------------------
// TDM: DMA a 2D tile [tile_d1 x tile_d0] (f32) from a row-major
// [tens_d1 x tens_d0] tensor at gaddr into LDS at lds_byte.
// D# bitfields per CDNA5 ISA ch.8 (fields straddle dword boundaries).
// Strides here fit in 32 bits (stride[47:32] = 0).
// ---------------------------------------------------------------------------
__device__ __forceinline__ void tdm_load_tile_f32(
    unsigned long long gaddr, unsigned int lds_byte,
    unsigned int tile_d0, unsigned int tile_d1,
    unsigned int tens_d0, unsigned int tens_d1,
    unsigned int stride0)
{
    v4u g0;
    g0.x = 1u;                                         // count=1, user D#
    g0.y = lds_byte;                                   // lds_addr[31:0]
    g0.z = (unsigned int)(gaddr & 0xffffffffull);      // global_addr[31:0]
    g0.w = (unsigned int)((gaddr >> 32) & 0x01fffffful)
         | 0x80000000u;                                // global_addr[56:32], type=2

    v8i g1;
    g1[0] = (int)(2u << 16);                           // data_size=4B (code 2), mask=0
    g1[1] = (int)((tens_d0 & 0xffffu) << 16);          // tensor_dim0[15:0] -> bits 63:48
    g1[2] = (int)(((tens_d0 >> 16) & 0xffffu)          // tensor_dim0[31:16]
         |        ((tens_d1 & 0xffffu) << 16));        // tensor_dim1[15:0]
    g1[3] = (int)(((tens_d1 >> 16) & 0xffffu)          // tensor_dim1[31:16]
         |        ((tile_d0 & 0xffffu) << 16));        // tile_dim0
    g1[4] = (int)(tile_d1 & 0xffffu);                  // tile_dim1 (tile_dim2=0)
    g1[5] = (int)stride0;                              // tensor_dim0_stride[31:0]
    g1[6] = 0;                                         // stride0[47:32]=0, dim1_stride lo=0
    g1[7] = 0;                                         // tensor_dim1_stride (unused, 2D)

    v4i g2 = {0, 0, 0, 0};
    v4i g3 = {0, 0, 0, 0};

#ifdef TDM_6ARG
    v8i g1b = {0, 0, 0, 0, 0, 0, 0, 0};
    __builtin_amdgcn_tensor_load_to_lds(g0, g1, g2, g3, g1b, 0);
#else
    __builtin_amdgcn_tensor_load_to_lds(g0, g1, g2, g3, 0);
#endif
}

// ---------------------------------------------------------------------------
// BN fold: scale = g * rsqrt(v + eps), shift = b - m * scale
// ---------------------------------------------------------------------------
__global__ __launch_bounds__(256) void bnfold_kernel(
    const float* __restrict__ g, const float* __restrict__ b,
    const float* __restrict__ m, const float* __restrict__ v,
    float* __restrict__ scale, float* __restrict__ shift, int C)
{
    int i = blockIdx.x * blockDim.x + threadIdx.x;
    if (i < C) {
        float s = g[i] * rsqrtf(v[i] + 1e-5f);
        scale[i] = s;
        shift[i] = b[i] - m[i] * s;
    }
}

// ---------------------------------------------------------------------------
// KNN top-3: one thread per query point. Key xyz tiles are DMA'd into LDS by
// the Tensor Data Mover (double-buffered, overlapping compute), synchronized
// with s_wait_tensorcnt + workgroup barrier.
// q: [B][3][N1]  k: [B][3][N2]  ->  idx/wgt: [B][N1][3]
// ---------------------------------------------------------------------------
__global__ __launch_bounds__(256) void knn3_kernel(
    const float* __restrict__ q, const float* __restrict__ kxyz,
    int* __restrict__ oidx, float* __restrict__ owgt)
{
    __shared__ float skey[2][3][KT2];   // [buf][xyz][key]

    const int b = blockIdx.y;
    const int n = blockIdx.x * blockDim.x + threadIdx.x;
    const bool issuer = (threadIdx.x < 32);   // wave 0 drives the TDM

    const float* qb = q + (size_t)b * 3 * N1_;
    const float qx = qb[n];
    const float qy = qb[N1_ + n];
    const float qz = qb[2 * N1_ + n];

    const float* kb = kxyz + (size_t)b * 3 * N2_;

    float d0 = 3.4e38f, d1 = 3.4e38f, d2 = 3.4e38f;
    int   i0 = 0,       i1 = 0,       i2 = 0;

    const int NT = N2_ / KT2;

    if (issuer) {
        tdm_load_tile_f32((unsigned long long)(uintptr_t)kb,
                          (unsigned int)(uintptr_t)&skey[0][0][0],
                          KT2, 3, N2_, 3, N2_);
    }

    for (int ti = 0; ti < NT; ++ti) {
        const int cur = ti & 1;
        if (issuer) {
            __builtin_amdgcn_s_wait_tensorcnt(0);   // current buffer landed
        }
        __syncthreads();                            // publish LDS tile to all waves
        if (issuer && (ti + 1) < NT) {              // prefetch next tile via TDM
            tdm_load_tile_f32((unsigned long long)(uintptr_t)(kb + (size_t)(ti + 1) * KT2),
                              (unsigned int)(uintptr_t)&skey[cur ^ 1][0][0],
                              KT2, 3, N2_, 3, N2_);
        }

        const float* sx = skey[cur][0];
        const float* sy = skey[cur][1];
        const float* sz = skey[cur][2];
        const int tbase = ti * KT2;

        #pragma unroll 4
        for (int j = 0; j < KT2; ++j) {
            float dx = qx - sx[j];
            float dy = qy - sy[j];
            float dz = qz - sz[j];
            float d  = fmaf(dx, dx, fmaf(dy, dy, dz * dz));
            int  gj  = tbase + j;
            if (d < d2) {
                if (d < d1) {
                    if (d < d0) { d2 = d1; i2 = i1; d1 = d0; i1 = i0; d0 = d; i0 = gj; }
                    else        { d2 = d1; i2 = i1; d1 = d;  i1 = gj; }
                } else          { d2 = d;  i2 = gj; }
            }
        }
        __syncthreads();   // all waves done reading buffer `cur` before it is re-DMA'd
    }

    float w0 = 1.0f / fmaxf(d0, 1e-10f);
    float w1 = 1.0f / fmaxf(d1, 1e-10f);
    float w2 = 1.0f / fmaxf(d2, 1e-10f);
    float inv = 1.0f / (w0 + w1 + w2);

    size_t o = ((size_t)b * N1_ + n) * 3;
    oidx[o] = i0; oidx[o + 1] = i1; oidx[o + 2] = i2;
    owgt[o] = w0 * inv; owgt[o + 1] = w1 * inv; owgt[o + 2] = w2 * inv;
}

// ---------------------------------------------------------------------------
// Interpolate (3-NN weighted gather) + concat skip features.
// Output X in K-pair interleaved layout: Xp[c/2][n] = {X[c][n], X[c+1][n]}
// (float2), so GEMM B-fragments are single b64 loads.
// feat: [B][256][N2], skip: [B][64][N1] -> Xp: [B][160][N1] float2
// ---------------------------------------------------------------------------
__global__ __launch_bounds__(256) void interp_concat_kernel(
    const float* __restrict__ feat, const float* __restrict__ skip,
    const int* __restrict__ idx, const float* __restrict__ wgt,
    float* __restrict__ X)
{
    int t = blockIdx.x * blockDim.x + threadIdx.x;
    int b = t >> 14;          // / N1_
    int n = t & (N1_ - 1);    // % N1_

    size_t o = ((size_t)b * N1_ + n) * 3;
    int   i0 = idx[o], i1 = idx[o + 1], i2 = idx[o + 2];
    float w0 = wgt[o], w1 = wgt[o + 1], w2 = wgt[o + 2];

    const float* fb = feat + (size_t)b * C_IN * N2_;
    float* Xb = X + (size_t)b * C0_ * N1_;   // interleaved: pair p at Xb + (p*N1 + n)*2

    #pragma unroll 4
    for (int c = 0; c < C_IN; c += 2) {
        const float* fr0 = fb + (size_t)c * N2_;
        const float* fr1 = fr0 + N2_;
        v2f p;
        p.x = w0 * fr0[i0] + w1 * fr0[i1] + w2 * fr0[i2];
        p.y = w0 * fr1[i0] + w1 * fr1[i1] + w2 * fr1[i2];
        *(v2f*)(Xb + ((size_t)(c >> 1) * N1_ + n) * 2) = p;
    }
    const float* sb = skip + (size_t)b * C_SKIP * N1_;
    #pragma unroll 4
    for (int c = 0; c < C_SKIP; c += 2) {
        v2f p;
        p.x = sb[(size_t)c * N1_ + n];
        p.y = sb[(size_t)(c + 1) * N1_ + n];
        *(v2f*)(Xb + ((size_t)((C_IN + c) >> 1) * N1_ + n) * 2) = p;
    }
}

// ---------------------------------------------------------------------------
// Fused GEMM + BN + ReLU using V_WMMA_F32_16X16X4_F32 (exact fp32 matrix op).
// Y[b][m][n] = relu(scale[m] * sum_k W[m][k] * X[b][k][n] + shift[m])
// X input is K-pair interleaved (float2 per (k/2, n)).
// OUT_ILV=true  -> write M-pair interleaved (feeds next GEMM layer, b64 stores)
// OUT_ILV=false -> write plain [B][M][N] (final output)
// Block: 256 threads = 8 waves arranged 2(M) x 4(N); wave tile 16M x 64N.
// Block tile: 32M x 256N.  grid = (N/256, M/32, B)
// ---------------------------------------------------------------------------
template <bool OUT_ILV>
__global__ __launch_bounds__(256) void gemm_bn_relu_kernel(
    const float* __restrict__ X,      // [B][K/2][N] float2 (interleaved)
    const float* __restrict__ W,      // [M][K] plain
    const float* __restrict__ scale,
    const float* __restrict__ shift,
    float* __restrict__ Y,
    int K, int M, int N)
{
    const int lane = threadIdx.x & 31;
    const int wave = threadIdx.x >> 5;
    const int hi   = lane >> 4;
    const int lx   = lane & 15;
    const int b    = blockIdx.z;

    const int m_base = (blockIdx.y * 2 + (wave >> 2)) * 16;
    const int n_base = blockIdx.x * 256 + (wave & 3) * 64;

    const float* Xb = X + (size_t)b * K * N;   // K/2 pair-rows of 2N floats

    v8f acc0 = {}, acc1 = {}, acc2 = {}, acc3 = {};

    for (int k = 0; k < K; k += 4) {
        // A fragment: (M = m_base+lx, K = k+2*hi, k+2*hi+1) -- b64 load
        const float* wr = W + (size_t)(m_base + lx) * K + (k + 2 * hi);
        v2f a = *(const v2f*)wr;

        // B fragments: pair-row (k+2*hi)/2, col n_base+16*j+lx -- b64 loads
        const float* xr = Xb + ((size_t)((k >> 1) + hi) * N + n_base + lx) * 2;
        __builtin_prefetch(xr + (size_t)4 * N, 0, 1);   // next k-step (global_prefetch_b8)

        v2f b0 = *(const v2f*)(xr);
        v2f b1 = *(const v2f*)(xr + 32);
        v2f b2 = *(const v2f*)(xr + 64);
        v2f b3 = *(const v2f*)(xr + 96);

        acc0 = __builtin_amdgcn_wmma_f32_16x16x4_f32(false, a, false, b0, (short)0, acc0, false, false);
        acc1 = __builtin_amdgcn_wmma_f32_16x16x4_f32(false, a, false, b1, (short)0, acc1, false, false);
        acc2 = __builtin_amdgcn_wmma_f32_16x16x4_f32(false, a, false, b2, (short)0, acc2, false, false);
        acc3 = __builtin_amdgcn_wmma_f32_16x16x4_f32(false, a, false, b3, (short)0, acc3, false, false);
    }

    // C layout: (VGPR r, lane) -> M = m_base + r + 8*hi, N = n_base + 16*j + lx.
    // Rows r and r+1 are consecutive M in the SAME lane -> pack b64 stores.
    if (OUT_ILV) {
        #pragma unroll
        for (int r = 0; r < 8; r += 2) {
            int m = m_base + r + 8 * hi;               // even
            float s0 = scale[m],     t0 = shift[m];
            float s1 = scale[m + 1], t1 = shift[m + 1];
            float* yr = Y + ((size_t)b * M * N) + ((size_t)(m >> 1) * N + n_base + lx) * 2;
            v2f p0; p0.x = fmaxf(fmaf(acc0[r], s0, t0), 0.0f); p0.y = fmaxf(fmaf(acc0[r + 1], s1, t1), 0.0f);
            v2f p1; p1.x = fmaxf(fmaf(acc1[r], s0, t0), 0.0f); p1.y = fmaxf(fmaf(acc1[r + 1], s1, t1), 0.0f);
            v2f p2; p2.x = fmaxf(fmaf(acc2[r], s0, t0), 0.0f); p2.y = fmaxf(fmaf(acc2[r + 1], s1, t1), 0.0f);
            v2f p3; p3.x = fmaxf(fmaf(acc3[r], s0, t0), 0.0f); p3.y = fmaxf(fmaf(acc3[r + 1], s1, t1), 0.0f);
            *(v2f*)(yr)      = p0;
            *(v2f*)(yr + 32) = p1;
            *(v2f*)(yr + 64) = p2;
            *(v2f*)(yr + 96) = p3;
        }
    } else {
        #pragma unroll
        for (int r = 0; r < 8; ++r) {
            int m = m_base + r + 8 * hi;
            float s = scale[m];
            float t = shift[m];
            size_t row = ((size_t)b * M + m) * N + n_base + lx;
            Y[row]      = fmaxf(fmaf(acc0[r], s, t), 0.0f);
            Y[row + 16] = fmaxf(fmaf(acc1[r], s, t), 0.0f);
            Y[row + 32] = fmaxf(fmaf(acc2[r], s, t), 0.0f);
            Y[row + 48] = fmaxf(fmaf(acc3[r], s, t), 0.0f);
        }
    }
}

// ---------------------------------------------------------------------------
// Host-side launcher
// ---------------------------------------------------------------------------
extern "C" void kernel_launch(void* const* d_in, const int* in_sizes, int n_in,
                              void* d_out, int out_size, void* d_ws, size_t ws_size,
                              hipStream_t stream) {
    (void)in_sizes; (void)n_in; (void)out_size; (void)ws_size;

    const float* in_xyz      = (const float*)d_in[0];   // [B,3,N2]
    const float* in_feature  = (const float*)d_in[1];   // [B,256,N2]
    const float* out_xyz     = (const float*)d_in[2];   // [B,3,N1]
    const float* out_feature = (const float*)d_in[3];   // [B,64,N1]
    const float* w1 = (const float*)d_in[4];
    const float* g1 = (const float*)d_in[5];
    const float* b1 = (const float*)d_in[6];
    const float* m1 = (const float*)d_in[7];
    const float* v1 = (const float*)d_in[8];
    const float* w2 = (const float*)d_in[9];
    const float* g2 = (const float*)d_in[10];
    const float* b2 = (const float*)d_in[11];
    const float* m2 = (const float*)d_in[12];
    const float* v2 = (const float*)d_in[13];

    // Workspace layout
    float* X   = (float*)d_ws;                                  // B*320*N1 (interleaved pairs)
    float* Y1  = X  + (size_t)B_ * C0_ * N1_;                   // B*256*N1 (interleaved pairs)
    int*   idx = (int*)(Y1 + (size_t)B_ * C1_ * N1_);           // B*N1*3
    float* wgt = (float*)(idx + (size_t)B_ * N1_ * 3);          // B*N1*3
    float* sc1 = wgt + (size_t)B_ * N1_ * 3;
    float* sh1 = sc1 + C1_;
    float* sc2 = sh1 + C1_;
    float* sh2 = sc2 + C1_;

    // 1. Fold BN params
    bnfold_kernel<<<1, 256, 0, stream>>>(g1, b1, m1, v1, sc1, sh1, C1_);
    bnfold_kernel<<<1, 256, 0, stream>>>(g2, b2, m2, v2, sc2, sh2, C1_);

    // 2. KNN top-3 (query = out_xyz, keys = in_xyz) — TDM-staged key tiles
    knn3_kernel<<<dim3(N1_ / 256, B_), 256, 0, stream>>>(out_xyz, in_xyz, idx, wgt);

    // 3. Interpolate + concat -> X (interleaved) [B,160,N1] float2
    interp_concat_kernel<<<(B_ * N1_) / 256, 256, 0, stream>>>(
        in_feature, out_feature, idx, wgt, X);

    // 4. Layer 1: Y1 = relu(bn(W1 @ X)), K=320, interleaved output
    gemm_bn_relu_kernel<true><<<dim3(N1_ / 256, C1_ / 32, B_), 256, 0, stream>>>(
        X, w1, sc1, sh1, Y1, C0_, C1_, N1_);

    // 5. Layer 2: out = relu(bn(W2 @ Y1)), K=256, plain output
    gemm_bn_relu_kernel<false><<<dim3(N1_ / 256, C1_ / 32, B_), 256, 0, stream>>>(
        Y1, w2, sc2, sh2, (float*)d_out, C1_, C1_, N1_);
}